// baselineNet_paper_Trans_34213709480228
// MI455X (gfx1250) — compile-verified
//
#include <hip/hip_runtime.h>

// ---------------------------------------------------------------------------
// Types for CDNA5 WMMA f32 16x16x4:  A is 16x4 (2 VGPRs), B is 4x16 (2 VGPRs),
// C/D is 16x16 f32 (8 VGPRs).
// ---------------------------------------------------------------------------
typedef __attribute__((ext_vector_type(2))) float v2f;
typedef __attribute__((ext_vector_type(8))) float v8f;

#define NEG_INF (-__builtin_inff())

#define N_NODES 1024
#define SEQ_LEN 1000
#define SEQ_HALF 500
#define LSTM_H 1000
#define GATES 4000
#define N_EDGES 8192
#define N_GRAPHS 64

// ---------------------------------------------------------------------------
// small helpers
// ---------------------------------------------------------------------------
__device__ inline void atomicMaxF(float* addr, float val) {
    unsigned int* ua = (unsigned int*)addr;
    unsigned int old = *ua;
    while (__uint_as_float(old) < val) {
        unsigned int prev = atomicCAS(ua, old, __float_as_uint(val));
        if (prev == old) break;
        old = prev;
    }
}

__global__ void fill_kernel(float* p, float v, int n) {
    int i = blockIdx.x * blockDim.x + threadIdx.x;
    if (i < n) p[i] = v;
}

// ---------------------------------------------------------------------------
// Fused conv1 -> bn -> relu -> maxpool(3,2,pad1) -> conv4 -> bn -> relu.
// One block per node. x: (1024,1000) -> seq: (1024,500)
// ---------------------------------------------------------------------------
__global__ void conv_front_kernel(const float* __restrict__ x,
                                  const float* __restrict__ w1, const float* __restrict__ b1,
                                  const float* __restrict__ g1, const float* __restrict__ bb1,
                                  const float* __restrict__ w4, const float* __restrict__ b4,
                                  const float* __restrict__ g4, const float* __restrict__ bb4,
                                  float* __restrict__ seq) {
    __shared__ float xr[SEQ_LEN];
    __shared__ float pl[4 * SEQ_HALF];
    const float inv = 0.999995000037499f;  // 1/sqrt(1+1e-5)
    int n = blockIdx.x;
    for (int i = threadIdx.x; i < SEQ_LEN; i += blockDim.x) xr[i] = x[n * SEQ_LEN + i];
    __syncthreads();
    // conv1 + bn1 + relu + maxpool -> pooled (4 x 500)
    for (int idx = threadIdx.x; idx < 4 * SEQ_HALF; idx += blockDim.x) {
        int ch = idx / SEQ_HALF, u = idx % SEQ_HALF;
        float mx = NEG_INF;
        for (int d = -1; d <= 1; ++d) {
            int s = 2 * u + d;
            if (s < 0 || s >= SEQ_LEN) continue;
            float v = 0.f;
            for (int kk = 0; kk < 3; ++kk) {
                int sx = s - 1 + kk;
                if (sx >= 0 && sx < SEQ_LEN) v += xr[sx] * w1[ch * 3 + kk];
            }
            v += b1[ch];
            v = g1[ch] * v * inv + bb1[ch];
            v = fmaxf(v, 0.f);
            mx = fmaxf(mx, v);
        }
        pl[ch * SEQ_HALF + u] = mx;
    }
    __syncthreads();
    // conv4 + bn4 + relu -> seq (500)
    for (int t = threadIdx.x; t < SEQ_HALF; t += blockDim.x) {
        float v = 0.f;
        for (int ch = 0; ch < 4; ++ch)
            for (int kk = 0; kk < 3; ++kk) {
                int u = t - 1 + kk;
                if (u >= 0 && u < SEQ_HALF) v += pl[ch * SEQ_HALF + u] * w4[ch * 3 + kk];
            }
        v += b4[0];
        v = g4[0] * v * inv + bb4[0];
        seq[n * SEQ_HALF + t] = fmaxf(v, 0.f);
    }
}

// ---------------------------------------------------------------------------
// f32 WMMA GEMM:  C[M,N] = act(A[M,K] @ B + bias0 + bias1)
//   BT=1: B stored (N,K) row-major  (x @ W.T — LSTM projections)
//   BT=0: B stored (K,N) row-major  (x @ W — MLPs)
// Block = 256 threads = 8 waves; each wave owns one 16x16 output tile.
// Fast path (tile fully in-bounds): unguarded loop, b64 loads for the
// K-contiguous operands; guarded path only for edge tiles / K tail.
// Fragment layout (wave32): lane l, li=l&15, hv=l>>4:
//   A vgpr v  = A[m0+li][k + v + 2*hv]
//   B vgpr v  = B[k + v + 2*hv][n0+li]
//   C vgpr r  = C[m0 + r + 8*hv][n0+li]
// ---------------------------------------------------------------------------
__global__ void gemm_wmma_kernel(const float* __restrict__ A, const float* __restrict__ B,
                                 const float* __restrict__ bias0, const float* __restrict__ bias1,
                                 float* __restrict__ C, int M, int N, int K, int BT, int act) {
    int lane = threadIdx.x & 31;
    int wv = threadIdx.x >> 5;
    int n0 = (blockIdx.x * 8 + wv) * 16;
    int m0 = blockIdx.y * 16;
    int hv = lane >> 4;
    int li = lane & 15;
    int mrow = m0 + li;
    int ncol = n0 + li;
    v8f acc = {};
    const bool full = (m0 + 16 <= M) && (n0 + 16 <= N);
    const int K4 = K & ~3;
    if (full) {
        // K is even for every full-tile GEMM here with K >= 4 -> v2f loads are
        // 8B aligned on the K-contiguous operands.
        const float* Ap = A + (size_t)mrow * K + 2 * hv;
        const float* Bp = BT ? (B + (size_t)ncol * K + 2 * hv)
                             : (B + (size_t)(2 * hv) * N + ncol);
        if (BT) {
#pragma unroll 4
            for (int k = 0; k < K4; k += 4) {
                v2f a = *(const v2f*)(Ap + k);
                v2f b = *(const v2f*)(Bp + k);
                acc = __builtin_amdgcn_wmma_f32_16x16x4_f32(false, a, false, b, (short)0, acc,
                                                            false, false);
            }
        } else {
#pragma unroll 4
            for (int k = 0; k < K4; k += 4) {
                v2f a = *(const v2f*)(Ap + k);
                v2f b;
                b.x = Bp[(size_t)k * N];
                b.y = Bp[(size_t)(k + 1) * N];
                acc = __builtin_amdgcn_wmma_f32_16x16x4_f32(false, a, false, b, (short)0, acc,
                                                            false, false);
            }
        }
        if (K4 < K) {  // guarded K tail (at most one step)
            int ka = K4 + 2 * hv;
            v2f a, b;
            a.x = (ka < K) ? A[(size_t)mrow * K + ka] : 0.f;
            a.y = (ka + 1 < K) ? A[(size_t)mrow * K + ka + 1] : 0.f;
            if (BT) {
                b.x = (ka < K) ? B[(size_t)ncol * K + ka] : 0.f;
                b.y = (ka + 1 < K) ? B[(size_t)ncol * K + ka + 1] : 0.f;
            } else {
                b.x = (ka < K) ? B[(size_t)ka * N + ncol] : 0.f;
                b.y = (ka + 1 < K) ? B[(size_t)(ka + 1) * N + ncol] : 0.f;
            }
            acc = __builtin_amdgcn_wmma_f32_16x16x4_f32(false, a, false, b, (short)0, acc,
                                                        false, false);
        }
    } else {
        for (int k = 0; k < K; k += 4) {
            int ka = k + 2 * hv;
            v2f a, b;
            a.x = (mrow < M && ka < K) ? A[(size_t)mrow * K + ka] : 0.f;
            a.y = (mrow < M && ka + 1 < K) ? A[(size_t)mrow * K + ka + 1] : 0.f;
            if (BT) {
                b.x = (ncol < N && ka < K) ? B[(size_t)ncol * K + ka] : 0.f;
                b.y = (ncol < N && ka + 1 < K) ? B[(size_t)ncol * K + ka + 1] : 0.f;
            } else {
                b.x = (ncol < N && ka < K) ? B[(size_t)ka * N + ncol] : 0.f;
                b.y = (ncol < N && ka + 1 < K) ? B[(size_t)(ka + 1) * N + ncol] : 0.f;
            }
            acc = __builtin_amdgcn_wmma_f32_16x16x4_f32(false, a, false, b, (short)0, acc,
                                                        false, false);
        }
    }
    float badd = 0.f;
    if (ncol < N) {
        if (bias0) badd += bias0[ncol];
        if (bias1) badd += bias1[ncol];
    }
    if (full) {
#pragma unroll
        for (int r = 0; r < 8; ++r) {
            float v = acc[r] + badd;
            if (act == 1) v = fmaxf(v, 0.f);
            else if (act == 2) v = 1.f / (1.f + __expf(-v));
            C[(size_t)(m0 + r + 8 * hv) * N + ncol] = v;
        }
    } else {
#pragma unroll
        for (int r = 0; r < 8; ++r) {
            int m = m0 + r + 8 * hv;
            if (m < M && ncol < N) {
                float v = acc[r] + badd;
                if (act == 1) v = fmaxf(v, 0.f);
                else if (act == 2) v = 1.f / (1.f + __expf(-v));
                C[(size_t)m * N + ncol] = v;
            }
        }
    }
}

// ---------------------------------------------------------------------------
// One LSTM recurrence step: gates = xrow + whh @ h_in ; update c, h.
// grid = 16 blocks x 256 threads: block b covers j in [b*64, b*64+64),
// thread tid handles gate g=tid>>6, column j.  h double-buffered across steps.
// h_in staged through LDS (reused by all 256 threads); whh rows read b128.
// ---------------------------------------------------------------------------
__global__ void lstm_step_kernel(const float* __restrict__ whh, const float* __restrict__ xrow,
                                 const float* __restrict__ hin, float* __restrict__ hout,
                                 float* __restrict__ cst, float* __restrict__ hsrow, int H) {
    __shared__ float hsh[LSTM_H];
    __shared__ float gl[256];
    int tid = threadIdx.x;
    for (int k = tid; k < H; k += blockDim.x) hsh[k] = hin[k];
    __syncthreads();
    int g = tid >> 6;
    int j = blockIdx.x * 64 + (tid & 63);
    float gate = 0.f;
    if (j < H) {
        size_t row = (size_t)(g * H + j);
        const float4* wr4 = (const float4*)(whh + row * H);  // H%4==0 -> 16B aligned
        __builtin_prefetch(wr4, 0, 0);
        float dot = 0.f;
        int H4 = H >> 2;
#pragma unroll 4
        for (int k = 0; k < H4; ++k) {
            float4 wq = wr4[k];
            dot = fmaf(wq.x, hsh[4 * k + 0], dot);
            dot = fmaf(wq.y, hsh[4 * k + 1], dot);
            dot = fmaf(wq.z, hsh[4 * k + 2], dot);
            dot = fmaf(wq.w, hsh[4 * k + 3], dot);
        }
        gate = dot + xrow[row];
    }
    gl[tid] = gate;
    __syncthreads();
    if (tid < 64) {
        int jj = blockIdx.x * 64 + tid;
        if (jj < H) {
            float ig = 1.f / (1.f + __expf(-gl[tid]));
            float fg = 1.f / (1.f + __expf(-gl[64 + tid]));
            float gg = tanhf(gl[128 + tid]);
            float og = 1.f / (1.f + __expf(-gl[192 + tid]));
            float cn = fg * cst[jj] + ig * gg;
            cst[jj] = cn;
            float hn = og * tanhf(cn);
            hout[jj] = hn;
            hsrow[jj] = hn;
        }
    }
}

// ---------------------------------------------------------------------------
// concat [a (M,Na) | b (M,Nb)] -> o (M,Na+Nb)
// ---------------------------------------------------------------------------
__global__ void concat_kernel(const float* __restrict__ a, const float* __restrict__ b,
                              float* __restrict__ o, int M, int Na, int Nb) {
    int idx = blockIdx.x * blockDim.x + threadIdx.x;
    int T = Na + Nb;
    if (idx >= M * T) return;
    int m = idx / T, jx = idx % T;
    o[idx] = (jx < Na) ? a[(size_t)m * Na + jx] : b[(size_t)m * Nb + (jx - Na)];
}

// ---------------------------------------------------------------------------
// transformer-conv edge kernels (heads=2)
// ---------------------------------------------------------------------------
__global__ void edge_score_kernel(const float* __restrict__ q, const float* __restrict__ k,
                                  const int* __restrict__ src, const int* __restrict__ dst,
                                  float* __restrict__ score, int nE, int heads, int C) {
    int idx = blockIdx.x * blockDim.x + threadIdx.x;
    if (idx >= nE * heads) return;
    int e = idx / heads, h = idx % heads;
    int hc = heads * C;
    const float* qd = q + (size_t)dst[e] * hc + h * C;
    const float* ks = k + (size_t)src[e] * hc + h * C;
    float s = 0.f;
    for (int cc = 0; cc < C; ++cc) s += qd[cc] * ks[cc];
    score[idx] = s / sqrtf((float)C);
}

__global__ void edge_segmax_kernel(const float* __restrict__ score, const int* __restrict__ dst,
                                   float* __restrict__ m, int nE, int heads) {
    int idx = blockIdx.x * blockDim.x + threadIdx.x;
    if (idx >= nE * heads) return;
    int e = idx / heads, h = idx % heads;
    atomicMaxF(&m[dst[e] * heads + h], score[idx]);
}

__global__ void edge_pexp_kernel(const float* __restrict__ score, const int* __restrict__ dst,
                                 const float* __restrict__ m, float* __restrict__ pexp,
                                 float* __restrict__ denom, int nE, int heads) {
    int idx = blockIdx.x * blockDim.x + threadIdx.x;
    if (idx >= nE * heads) return;
    int e = idx / heads, h = idx % heads;
    float p = __expf(score[idx] - m[dst[e] * heads + h]);
    pexp[idx] = p;
    atomicAdd(&denom[dst[e] * heads + h], p);
}

__global__ void edge_agg_kernel(const float* __restrict__ pexp, const float* __restrict__ denom,
                                const float* __restrict__ v, const int* __restrict__ src,
                                const int* __restrict__ dst, float* __restrict__ agg,
                                int nE, int heads, int C) {
    int hc = heads * C;
    int idx = blockIdx.x * blockDim.x + threadIdx.x;
    if (idx >= nE * hc) return;
    int e = idx / hc, cc = idx % hc;
    int h = cc / C;
    float alpha = pexp[e * heads + h] / denom[dst[e] * heads + h];
    atomicAdd(&agg[(size_t)dst[e] * hc + cc], alpha * v[(size_t)src[e] * hc + cc]);
}

__global__ void tconv_combine_kernel(const float* __restrict__ agg, const float* __restrict__ xs,
                                     float* __restrict__ out, int N, int heads, int C) {
    int idx = blockIdx.x * blockDim.x + threadIdx.x;
    if (idx >= N * C) return;
    int n = idx / C, cc = idx % C;
    int hc = heads * C;
    float s = 0.f;
    for (int h = 0; h < heads; ++h) s += agg[(size_t)n * hc + h * C + cc];
    out[idx] = s / (float)heads + xs[idx];
}

__global__ void add3_kernel(const float* a, const float* b, const float* c, float* o, int n) {
    int i = blockIdx.x * blockDim.x + threadIdx.x;
    if (i < n) o[i] = a[i] + b[i] + c[i];
}

__global__ void pool_max_kernel(const float* __restrict__ s, const int* __restrict__ batch,
                                float* __restrict__ pooled, int nodes, int C) {
    int idx = blockIdx.x * blockDim.x + threadIdx.x;
    if (idx >= nodes * C) return;
    int n = idx / C, cc = idx % C;
    atomicMaxF(&pooled[batch[n] * C + cc], s[idx]);
}

// ---------------------------------------------------------------------------
// host orchestration
// ---------------------------------------------------------------------------
struct Pp {
    const float *conv1_w, *conv1_b, *bn1_g, *bn1_b, *conv4_w, *conv4_b, *bn4_g, *bn4_b;
    const float *wih0, *whh0, *bih0, *bhh0, *wih1, *whh1, *bih1, *bhh1;
    const float *sta_w1, *sta_b1, *sta_w2, *sta_b2;
    const float *tot_w1, *tot_b1, *tot_w2, *tot_b2;
    const float *mlp1_w1, *mlp1_b1, *mlp1_w2, *mlp1_b2;
    const float *g_wq[4], *g_bq[4], *g_wk[4], *g_bk[4], *g_wv[4], *g_bv[4], *g_ws[4], *g_bs[4];
};

extern "C" void kernel_launch(void* const* d_in, const int* in_sizes, int n_in,
                              void* d_out, int out_size, void* d_ws, size_t ws_size,
                              hipStream_t stream) {
    auto F = [&](int i) { return (const float*)d_in[i]; };
    const float* x = F(0);
    const int* ei = (const int*)d_in[1];
    const int* batch = (const int*)d_in[2];
    const float* station = F(3);
    const int* src = ei;
    const int* dst = ei + N_EDGES;

    Pp p;
    bool insertion = (n_in > 4 && in_sizes[4] == 12);  // conv1_w first?
    if (insertion) {
        p.conv1_w = F(4);  p.conv1_b = F(5);  p.bn1_g = F(6);  p.bn1_b = F(7);
        p.conv4_w = F(8);  p.conv4_b = F(9);  p.bn4_g = F(10); p.bn4_b = F(11);
        p.wih0 = F(12); p.whh0 = F(13); p.bih0 = F(14); p.bhh0 = F(15);
        p.wih1 = F(16); p.whh1 = F(17); p.bih1 = F(18); p.bhh1 = F(19);
        p.sta_w1 = F(20); p.sta_b1 = F(21); p.sta_w2 = F(22); p.sta_b2 = F(23);
        p.tot_w1 = F(24); p.tot_b1 = F(25); p.tot_w2 = F(26); p.tot_b2 = F(27);
        p.mlp1_w1 = F(28); p.mlp1_b1 = F(29); p.mlp1_w2 = F(30); p.mlp1_b2 = F(31);
        for (int L = 0; L < 4; ++L) {
            int b = 32 + L * 8;
            p.g_wq[L] = F(b + 0); p.g_bq[L] = F(b + 1);
            p.g_wk[L] = F(b + 2); p.g_bk[L] = F(b + 3);
            p.g_wv[L] = F(b + 4); p.g_bv[L] = F(b + 5);
            p.g_ws[L] = F(b + 6); p.g_bs[L] = F(b + 7);
        }
    } else {  // sorted-key pytree order
        p.bn1_b = F(4);  p.bn1_g = F(5);  p.bn4_b = F(6);  p.bn4_g = F(7);
        p.conv1_b = F(8); p.conv1_w = F(9); p.conv4_b = F(10); p.conv4_w = F(11);
        for (int L = 0; L < 4; ++L) {
            int b = 12 + L * 8;
            p.g_bk[L] = F(b + 0); p.g_bq[L] = F(b + 1);
            p.g_bs[L] = F(b + 2); p.g_bv[L] = F(b + 3);
            p.g_wk[L] = F(b + 4); p.g_wq[L] = F(b + 5);
            p.g_ws[L] = F(b + 6); p.g_wv[L] = F(b + 7);
        }
        p.bhh0 = F(44); p.bih0 = F(45); p.whh0 = F(46); p.wih0 = F(47);
        p.bhh1 = F(48); p.bih1 = F(49); p.whh1 = F(50); p.wih1 = F(51);
        p.mlp1_b1 = F(52); p.mlp1_b2 = F(53); p.mlp1_w1 = F(54); p.mlp1_w2 = F(55);
        p.sta_b1 = F(56); p.sta_b2 = F(57); p.sta_w1 = F(58); p.sta_w2 = F(59);
        p.tot_b1 = F(60); p.tot_b2 = F(61); p.tot_w1 = F(62); p.tot_w2 = F(63);
    }

    // ---- workspace layout ----
    float* w = (float*)d_ws;
    auto alloc = [&](size_t n) { float* q = w; w += n; return q; };
    float* SEQF  = alloc((size_t)N_NODES * SEQ_HALF);
    float* XPROJ = alloc((size_t)N_NODES * GATES);
    float* HS0   = alloc((size_t)N_NODES * LSTM_H);
    float* HS1   = alloc((size_t)N_NODES * LSTM_H);
    float* HA = alloc(1024); float* HB = alloc(1024); float* CB = alloc(1024);
    float* STAH  = alloc((size_t)N_NODES * 128);
    float* STAO  = alloc((size_t)N_NODES * 250);
    float* TOTIN = alloc((size_t)N_NODES * 1250);
    float* TOTH  = alloc((size_t)N_NODES * 512);
    float* FEAT  = alloc((size_t)N_NODES * 256);
    float* QB = alloc((size_t)N_NODES * 256);
    float* KB = alloc((size_t)N_NODES * 256);
    float* VB = alloc((size_t)N_NODES * 256);
    float* XSB = alloc((size_t)N_NODES * 128);
    float* SCORE = alloc((size_t)N_EDGES * 2);
    float* MB  = alloc((size_t)N_NODES * 2);
    float* DEN = alloc((size_t)N_NODES * 2);
    float* PEXP = alloc((size_t)N_EDGES * 2);
    float* AGG = alloc((size_t)N_NODES * 256);
    float* X0 = alloc((size_t)N_NODES * 128);
    float* X1 = alloc((size_t)N_NODES * 64);
    float* X2 = alloc((size_t)N_NODES * 64);
    float* X3 = alloc((size_t)N_NODES * 64);
    float* SUMB = alloc((size_t)N_NODES * 64);
    float* POOLB = alloc((size_t)N_GRAPHS * 64);
    float* MH = alloc((size_t)N_GRAPHS * 16);

    auto nb = [](int n) { return (n + 255) / 256; };
    auto fill = [&](float* q, float v, int n) {
        fill_kernel<<<nb(n), 256, 0, stream>>>(q, v, n);
    };
    auto gemm = [&](const float* A, const float* B, const float* b0, const float* b1,
                    float* C, int M, int N, int K, int BT, int act) {
        dim3 g((N + 127) / 128, (M + 15) / 16);
        gemm_wmma_kernel<<<g, 256, 0, stream>>>(A, B, b0, b1, C, M, N, K, BT, act);
    };

    // ---- 1. conv front-end ----
    conv_front_kernel<<<N_NODES, 256, 0, stream>>>(x, p.conv1_w, p.conv1_b, p.bn1_g, p.bn1_b,
                                                   p.conv4_w, p.conv4_b, p.bn4_g, p.bn4_b, SEQF);

    // ---- 2. two LSTM layers (scan over the 1024 nodes) ----
    auto lstm_layer = [&](const float* xin, int K, const float* wih, const float* whh,
                          const float* bih, const float* bhh, float* hs) {
        gemm(xin, wih, bih, bhh, XPROJ, N_NODES, GATES, K, 1, 0);  // x @ wih.T + bih + bhh
        fill(HA, 0.f, 1024);
        fill(CB, 0.f, 1024);
        float* ha = HA;
        float* hb = HB;
        for (int t = 0; t < N_NODES; ++t) {
            lstm_step_kernel<<<16, 256, 0, stream>>>(whh, XPROJ + (size_t)t * GATES, ha, hb, CB,
                                                     hs + (size_t)t * LSTM_H, LSTM_H);
            float* tmp = ha; ha = hb; hb = tmp;
        }
    };
    lstm_layer(SEQF, SEQ_HALF, p.wih0, p.whh0, p.bih0, p.bhh0, HS0);
    lstm_layer(HS0, LSTM_H, p.wih1, p.whh1, p.bih1, p.bhh1, HS1);

    // ---- 3. sta MLP + concat + tot MLP ----
    gemm(station, p.sta_w1, p.sta_b1, nullptr, STAH, N_NODES, 128, 3, 0, 1);
    gemm(STAH, p.sta_w2, p.sta_b2, nullptr, STAO, N_NODES, 250, 128, 0, 0);
    concat_kernel<<<nb(N_NODES * 1250), 256, 0, stream>>>(HS1, STAO, TOTIN, N_NODES, 1000, 250);
    gemm(TOTIN, p.tot_w1, p.tot_b1, nullptr, TOTH, N_NODES, 512, 1250, 0, 1);
    gemm(TOTH, p.tot_w2, p.tot_b2, nullptr, FEAT, N_NODES, 256, 512, 0, 0);

    // ---- 4. four transformer-conv graph layers ----
    auto tconv = [&](const float* xin, int in, int c, int L, float* out) {
        int hc = 2 * c;
        gemm(xin, p.g_wq[L], p.g_bq[L], nullptr, QB, N_NODES, hc, in, 0, 0);
        gemm(xin, p.g_wk[L], p.g_bk[L], nullptr, KB, N_NODES, hc, in, 0, 0);
        gemm(xin, p.g_wv[L], p.g_bv[L], nullptr, VB, N_NODES, hc, in, 0, 0);
        gemm(xin, p.g_ws[L], p.g_bs[L], nullptr, XSB, N_NODES, c, in, 0, 0);
        fill(MB, NEG_INF, N_NODES * 2);
        fill(DEN, 0.f, N_NODES * 2);
        fill(AGG, 0.f, N_NODES * hc);
        int tE = N_EDGES * 2;
        edge_score_kernel<<<nb(tE), 256, 0, stream>>>(QB, KB, src, dst, SCORE, N_EDGES, 2, c);
        edge_segmax_kernel<<<nb(tE), 256, 0, stream>>>(SCORE, dst, MB, N_EDGES, 2);
        edge_pexp_kernel<<<nb(tE), 256, 0, stream>>>(SCORE, dst, MB, PEXP, DEN, N_EDGES, 2);
        int tA = N_EDGES * hc;
        edge_agg_kernel<<<nb(tA), 256, 0, stream>>>(PEXP, DEN, VB, src, dst, AGG, N_EDGES, 2, c);
        tconv_combine_kernel<<<nb(N_NODES * c), 256, 0, stream>>>(AGG, XSB, out, N_NODES, 2, c);
    };
    tconv(FEAT, 256, 128, 0, X0);
    tconv(X0, 128, 64, 1, X1);
    tconv(X1, 64, 64, 2, X2);
    tconv(X2, 64, 64, 3, X3);

    // ---- 5. graph max-pool + mlp1 + sigmoid ----
    add3_kernel<<<nb(N_NODES * 64), 256, 0, stream>>>(X1, X2, X3, SUMB, N_NODES * 64);
    fill(POOLB, NEG_INF, N_GRAPHS * 64);
    pool_max_kernel<<<nb(N_NODES * 64), 256, 0, stream>>>(SUMB, batch, POOLB, N_NODES, 64);
    gemm(POOLB, p.mlp1_w1, p.mlp1_b1, nullptr, MH, N_GRAPHS, 16, 64, 0, 1);
    gemm(MH, p.mlp1_w2, p.mlp1_b2, nullptr, (float*)d_out, N_GRAPHS, 3, 16, 0, 2);
    (void)ws_size; (void)out_size; (void)n_in;
}